// HierarchicalClassificationLoss_10376640987955
// MI455X (gfx1250) — compile-verified
//
#include <hip/hip_runtime.h>
#include <math.h>

#define BATCH 65536
#define NCLS 2048
#define ROWS_PER_BLK 16
#define THREADS 256
#define NBLK (BATCH / ROWS_PER_BLK)   // 4096
#define PSTRIDE 2052                  // padded LDS row stride (floats): 2052 % 64 == 4 -> conflict-free
#define TDIM 32

typedef __attribute__((ext_vector_type(2))) float v2f;
typedef __attribute__((ext_vector_type(8))) float v8f;

// ---------------- Kernel 0: tiled transpose H -> HT (optional, if workspace is big enough) ----------------
__global__ void __launch_bounds__(256)
transpose_kernel(const float* __restrict__ H, float* __restrict__ HT) {
    __shared__ float tile[TDIM][TDIM + 1];
    const int ntx = NCLS / TDIM;                 // 64 tiles per side
    const int bx = blockIdx.x % ntx;             // column tile
    const int by = blockIdx.x / ntx;             // row tile
    const int tx  = threadIdx.x & (TDIM - 1);    // 0..31
    const int ty0 = threadIdx.x >> 5;            // 0..7
    #pragma unroll
    for (int ty = ty0; ty < TDIM; ty += 8)
        tile[ty][tx] = H[(size_t)(by * TDIM + ty) * NCLS + bx * TDIM + tx];
    __syncthreads();
    #pragma unroll
    for (int ty = ty0; ty < TDIM; ty += 8)
        HT[(size_t)(bx * TDIM + ty) * NCLS + by * TDIM + tx] = tile[tx][ty];
}

// ---------------- Kernel 1: row sums (has_child) and column sums (has_parent) of H ----------------
__global__ void __launch_bounds__(THREADS)
hier_sums_kernel(const float* __restrict__ H, float* __restrict__ colsum, float* __restrict__ rowsum) {
    __shared__ float sr[THREADS];
    __shared__ float sc[THREADS];
    const int j = blockIdx.x;
    const int tid = threadIdx.x;
    float rs = 0.f, cs = 0.f;
    for (int c = tid; c < NCLS; c += THREADS) {
        rs += H[(size_t)j * NCLS + c];   // row j, coalesced
        cs += H[(size_t)c * NCLS + j];   // column j, strided (one-time, L2 resident)
    }
    sr[tid] = rs; sc[tid] = cs;
    __syncthreads();
    for (int s = THREADS / 2; s > 0; s >>= 1) {
        if (tid < s) { sr[tid] += sr[tid + s]; sc[tid] += sc[tid + s]; }
        __syncthreads();
    }
    if (tid == 0) { rowsum[j] = sr[0]; colsum[j] = sc[0]; }
}

// ---------------- Kernel 2: main fused softmax + WMMA dot products ----------------
// USE_HT=true: parent operand streamed from pre-transposed HT (contiguous b64 per lane).
// USE_HT=false: parent operand gathered as a column of H (two scattered b32 per lane).
template <bool USE_HT>
__global__ void __launch_bounds__(THREADS)
hcl_main_kernel(const float* __restrict__ logits, const int* __restrict__ targets,
                const float* __restrict__ H, const float* __restrict__ HT,
                const float* __restrict__ colsum, const float* __restrict__ rowsum,
                float* __restrict__ partials) {
    extern __shared__ float smem[];
    float* pE     = smem;                          // 16 * PSTRIDE : unnormalized exp(l - max)
    float* rowDen = pE + ROWS_PER_BLK * PSTRIDE;   // 16
    float* diagP  = rowDen + 16;                   // 8 * 16 partial parent diagonals
    float* diagC  = diagP + 128;                   // 8 * 16 partial child diagonals
    float* hv     = diagC + 128;                   // 16 per-row loss values
    int*   tgt    = (int*)(hv + 16);               // 16 targets

    const int tid  = threadIdx.x;
    const int lane = tid & 31;
    const int w    = tid >> 5;                     // wave id, 0..7
    const int r0   = blockIdx.x * ROWS_PER_BLK;

    if (tid < ROWS_PER_BLK) tgt[tid] = targets[r0 + tid];

    // ---- Phase 1: each wave owns 2 rows; compute max, exp(l-max) -> LDS, and denom ----
    for (int rr = 0; rr < 2; ++rr) {
        const int m = w * 2 + rr;
        const float4* src = (const float4*)(logits + (size_t)(r0 + m) * NCLS);
        float4 vals[16];
        float lmax = -3.402823466e38f;
        #pragma unroll
        for (int i = 0; i < 16; ++i) {
            vals[i] = src[i * 32 + lane];
            lmax = fmaxf(lmax, fmaxf(fmaxf(vals[i].x, vals[i].y), fmaxf(vals[i].z, vals[i].w)));
        }
        #pragma unroll
        for (int off = 16; off > 0; off >>= 1)
            lmax = fmaxf(lmax, __shfl_xor(lmax, off, 32));
        float lsum = 0.f;
        float* dstRow = pE + m * PSTRIDE;          // 16B-aligned (PSTRIDE multiple of 4 floats)
        #pragma unroll
        for (int i = 0; i < 16; ++i) {
            float4 v = vals[i], e;
            e.x = __expf(v.x - lmax); e.y = __expf(v.y - lmax);
            e.z = __expf(v.z - lmax); e.w = __expf(v.w - lmax);
            lsum += (e.x + e.y) + (e.z + e.w);
            ((float4*)dstRow)[i * 32 + lane] = e;
        }
        #pragma unroll
        for (int off = 16; off > 0; off >>= 1)
            lsum += __shfl_xor(lsum, off, 32);
        if (lane == 0) rowDen[m] = lsum;
    }
    __syncthreads();

    // ---- Phase 2: WMMA f32 16x16x4. Each wave covers K in [w*256, w*256+256).
    // D_parent = E @ Bp where Bp[:,j] = H[:, t_j] = HT[t_j, :];  D_child = E @ Bc where Bc[:,j] = H[t_j, :].
    // Only the diagonals are needed.
    v8f cp = {0.f, 0.f, 0.f, 0.f, 0.f, 0.f, 0.f, 0.f};
    v8f cc = {0.f, 0.f, 0.f, 0.f, 0.f, 0.f, 0.f, 0.f};
    const int mrow = lane & 15;                    // M (A) / N (B) index for this lane
    const int koff = (lane >> 4) << 1;             // lanes 0-15 -> K {0,1}; lanes 16-31 -> K {2,3}
    const int tj   = tgt[mrow];
    const float* aRow = pE + mrow * PSTRIDE + koff;
    const float* hRow = H + (size_t)tj * NCLS + koff;                       // child: row of H
    const float* pRow = (USE_HT ? HT : H) + (size_t)tj * NCLS + koff;       // parent: row of HT
    const int kbase = w * 256;
    for (int k = 0; k < 256; k += 4) {
        const int kk = kbase + k;
        v2f a  = *(const v2f*)(aRow + kk);                       // LDS, conflict-free b64
        v2f bc = *(const v2f*)(hRow + kk);                       // global b64, contiguous
        v2f bp;
        if (USE_HT) {
            bp = *(const v2f*)(pRow + kk);                       // global b64, contiguous
        } else {
            bp.x = H[(size_t)(kk + koff)     * NCLS + tj];       // column gather fallback
            bp.y = H[(size_t)(kk + koff + 1) * NCLS + tj];
        }
        cp = __builtin_amdgcn_wmma_f32_16x16x4_f32(false, a, false, bp, (short)0, cp, false, false);
        cc = __builtin_amdgcn_wmma_f32_16x16x4_f32(false, a, false, bc, (short)0, cc, false, false);
    }

    // ---- Extract diagonal: C/D layout: VGPR d, lanes 0-15 => (M=d, N=lane); lanes 16-31 => (M=8+d, N=lane-16)
    #pragma unroll
    for (int d = 0; d < 8; ++d) {
        if (lane == d)      { diagP[w * 16 + d]     = cp[d]; diagC[w * 16 + d]     = cc[d]; }
        if (lane == 24 + d) { diagP[w * 16 + d + 8] = cp[d]; diagC[w * 16 + d + 8] = cc[d]; }
    }
    __syncthreads();

    // ---- Phase 3: per-row scalars (first 16 threads) ----
    if (tid < ROWS_PER_BLK) {
        const int j = tid;
        const int t = tgt[j];
        const float den = rowDen[j];
        const float et  = pE[j * PSTRIDE + t];
        float ps = 0.f, cs = 0.f;
        #pragma unroll
        for (int ww = 0; ww < 8; ++ww) { ps += diagP[ww * 16 + j]; cs += diagC[ww * 16 + j]; }
        const float inv = 1.f / den;
        const float pt = et * inv;
        ps *= inv; cs *= inv;
        const float ce = logf(den) - logf(et);     // -(log_softmax at target)
        float h = 0.f;
        if (colsum[t] > 0.f) h += fmaxf(pt - ps, 0.f);
        if (rowsum[t] > 0.f) h += fmaxf(cs - pt, 0.f);
        hv[j] = ce + h;
    }
    __syncthreads();
    if (tid == 0) {
        float s = 0.f;
        #pragma unroll
        for (int j = 0; j < ROWS_PER_BLK; ++j) s += hv[j];
        partials[blockIdx.x] = s;                  // deterministic per-block partial
    }
}

// ---------------- Kernel 3: deterministic final reduction ----------------
__global__ void __launch_bounds__(256)
final_reduce_kernel(const float* __restrict__ partials, float* __restrict__ out) {
    __shared__ float s[256];
    const int tid = threadIdx.x;
    float acc = 0.f;
    for (int i = tid; i < NBLK; i += 256) acc += partials[i];
    s[tid] = acc;
    __syncthreads();
    for (int st = 128; st > 0; st >>= 1) {
        if (tid < st) s[tid] += s[tid + st];
        __syncthreads();
    }
    if (tid == 0) out[0] = s[0] * (1.0f / (float)BATCH);   // (ce_sum + h_sum) / n
}

extern "C" void kernel_launch(void* const* d_in, const int* in_sizes, int n_in,
                              void* d_out, int out_size, void* d_ws, size_t ws_size,
                              hipStream_t stream) {
    (void)in_sizes; (void)n_in; (void)out_size;
    const float* logits  = (const float*)d_in[0];
    const int*   targets = (const int*)d_in[1];
    const float* H       = (const float*)d_in[2];

    float* ws       = (float*)d_ws;
    float* partials = ws;                   // NBLK floats
    float* colsum   = ws + NBLK;            // NCLS floats
    float* rowsum   = ws + NBLK + NCLS;     // NCLS floats
    float* HT       = ws + NBLK + 2 * NCLS; // NCLS*NCLS floats (optional, 16 MB)

    const size_t needHT = (size_t)(NBLK + 2 * NCLS + NCLS * NCLS) * sizeof(float);
    const bool useHT = (ws_size >= needHT);

    if (useHT) {
        transpose_kernel<<<(NCLS / TDIM) * (NCLS / TDIM), 256, 0, stream>>>(H, HT);
    }
    hier_sums_kernel<<<NCLS, THREADS, 0, stream>>>(H, colsum, rowsum);

    const size_t smemBytes =
        (ROWS_PER_BLK * PSTRIDE + 16 + 128 + 128 + 16) * sizeof(float) + ROWS_PER_BLK * sizeof(int);
    if (useHT) {
        hcl_main_kernel<true><<<NBLK, THREADS, smemBytes, stream>>>(
            logits, targets, H, HT, colsum, rowsum, partials);
    } else {
        hcl_main_kernel<false><<<NBLK, THREADS, smemBytes, stream>>>(
            logits, targets, H, HT, colsum, rowsum, partials);
    }

    final_reduce_kernel<<<1, 256, 0, stream>>>(partials, (float*)d_out);
}